// GAT_54546084659384
// MI455X (gfx1250) — compile-verified
//
#include <hip/hip_runtime.h>
#include <hip/hip_bf16.h>

typedef float v2f __attribute__((ext_vector_type(2)));
typedef float v8f __attribute__((ext_vector_type(8)));
typedef int   v4i __attribute__((vector_size(16)));

#define B_    1024
#define NNODE 16
#define FIN   1024
#define FOUT  4096
#define KBIG  (NNODE * FIN)   // 16384
#define MROWS (B_ * NNODE)    // 16384

// ---------------------------------------------------------------------------
// Async global->LDS copy path (gfx1250). Guarded so compile never breaks.
// Builtin signature (from clang diagnostic): (v4i AS1*, v4i AS3*, Imm, Imm)
// ---------------------------------------------------------------------------
#if defined(__has_builtin)
#if __has_builtin(__builtin_amdgcn_global_load_async_to_lds_b128) && \
    __has_builtin(__builtin_amdgcn_s_wait_asynccnt)
#define ASYNC_OK 1
#endif
#endif

typedef __attribute__((address_space(1))) v4i* g4_t;
typedef __attribute__((address_space(3))) v4i* s4_t;

__device__ __forceinline__ void cp16(const float* g, float* l)
{
#ifdef ASYNC_OK
    __builtin_amdgcn_global_load_async_to_lds_b128(
        (g4_t)(v4i*)(float*)g, (s4_t)(v4i*)l, 0, 0);
#else
    *(float4*)l = *(const float4*)g;
#endif
}

#ifdef ASYNC_OK
#define WAIT_ASYNC(n) __builtin_amdgcn_s_wait_asynccnt(n)
#else
#define WAIT_ASYNC(n)
#endif

// ---------------------------------------------------------------------------
// Kernel 1: wg[i][j] = sum_k X[i][k] * W1[j][k] + b1[j]
// X: [16384,1024], W1: [1024,1024] row-major, wg -> ws.
// Block tile 128(M) x 64(N), K-panel 16, double-buffered async LDS staging.
// 8 waves (4x2); each wave 32x32 = 2x2 WMMA f32 16x16 tiles.
// LDS tiles stored [row][k] with 20-float row stride (conflict-free v2f frags).
// ---------------------------------------------------------------------------
__global__ __launch_bounds__(256)
void k1_gemm_wg(const float* __restrict__ X, const float* __restrict__ W1,
                const float* __restrict__ b1, float* __restrict__ wg)
{
    __shared__ float As[2][128][20];   // [buf][m][k]
    __shared__ float Bs[2][64][20];    // [buf][n][k]

    const int t    = threadIdx.x;
    const int lane = t & 31;
    const int wid  = t >> 5;
    const int wm   = wid >> 1;     // 0..3 : M group of 32
    const int wn   = wid & 1;      // 0..1 : N group of 32
    const int lm   = lane & 15;
    const int lh   = lane >> 4;
    const int iBase = blockIdx.y * 128;
    const int jBase = blockIdx.x * 64;

    v8f acc[2][2] = {};

    const int am  = (t >> 2);      // A: chunk -> (m, kq); 2 chunks/thread
    const int akq = (t & 3);

    // issue K-panel 0 into buffer 0 (3 async b128 per thread)
    #pragma unroll
    for (int rep = 0; rep < 2; ++rep) {
        int m = am + rep * 64;
        cp16(&X[(size_t)(iBase + m) * FIN + akq * 4], &As[0][m][akq * 4]);
    }
    cp16(&W1[(size_t)(jBase + (t >> 2)) * FIN + (t & 3) * 4], &Bs[0][t >> 2][(t & 3) * 4]);

    int ib = 0;
    for (int kt = 0; kt < FIN; kt += 16, ib ^= 1) {
        if (kt + 16 < FIN) {
            // prefetch-issue next panel into other buffer
            #pragma unroll
            for (int rep = 0; rep < 2; ++rep) {
                int m = am + rep * 64;
                cp16(&X[(size_t)(iBase + m) * FIN + kt + 16 + akq * 4],
                     &As[ib ^ 1][m][akq * 4]);
            }
            cp16(&W1[(size_t)(jBase + (t >> 2)) * FIN + kt + 16 + (t & 3) * 4],
                 &Bs[ib ^ 1][t >> 2][(t & 3) * 4]);
            WAIT_ASYNC(3);   // in-order retirement: current panel landed
        } else {
            WAIT_ASYNC(0);
        }
        __syncthreads();

        #pragma unroll
        for (int kk = 0; kk < 16; kk += 4) {
            v2f af[2], bf[2];
            #pragma unroll
            for (int mt = 0; mt < 2; ++mt)
                af[mt] = *(const v2f*)&As[ib][wm * 32 + mt * 16 + lm][kk + 2 * lh];
            #pragma unroll
            for (int nt = 0; nt < 2; ++nt)
                bf[nt] = *(const v2f*)&Bs[ib][wn * 32 + nt * 16 + lm][kk + 2 * lh];
            #pragma unroll
            for (int mt = 0; mt < 2; ++mt)
                #pragma unroll
                for (int nt = 0; nt < 2; ++nt)
                    acc[mt][nt] = __builtin_amdgcn_wmma_f32_16x16x4_f32(
                        false, af[mt], false, bf[nt],
                        (short)0, acc[mt][nt], false, false);
        }
        __syncthreads();
    }

    // epilogue: + b1, store wg
    #pragma unroll
    for (int mt = 0; mt < 2; ++mt)
        #pragma unroll
        for (int nt = 0; nt < 2; ++nt) {
            int col = jBase + wn * 32 + nt * 16 + lm;
            float bias = b1[col];
            #pragma unroll
            for (int r = 0; r < 8; ++r) {
                int row = iBase + wm * 32 + mt * 16 + 8 * lh + r;
                wg[(size_t)row * FIN + col] = acc[mt][nt][r] + bias;
            }
        }
}

// ---------------------------------------------------------------------------
// Kernel 2: per-batch attention math. One block per batch, 256 threads.
// Thread (i,j): i = t/16, j = t%16.
// ---------------------------------------------------------------------------
__global__ __launch_bounds__(256)
void k2_attention(const float* __restrict__ X, const float* __restrict__ Mtx,
                  const float* __restrict__ a_param,
                  float* __restrict__ wg_attn,        // ws: wg in, attn out
                  float* __restrict__ e2_out, float* __restrict__ m_out)
{
    __shared__ float xs[16][257];
    __shared__ float gs[16][257];
    __shared__ float red[16][17];
    __shared__ float EM[16][17];   // pre-softmax scores (a_scores)
    __shared__ float ESM[16][17];  // softmaxed
    __shared__ float FI[16][17];
    __shared__ float FJ[16][17];
    __shared__ float cmax[16], csum[16], nX[16], nG[16];

    const int t = threadIdx.x;
    const int i = t >> 4, j = t & 15;
    const int b = blockIdx.x;
    const float aval = a_param[0];
    const float* xb = X + (size_t)b * NNODE * FIN;
    float* wb = wg_attn + (size_t)b * NNODE * FIN;
    const float* mb = Mtx + (size_t)b * NNODE * NNODE;

    float S = 0.f, nxp = 0.f, ngp = 0.f;
    for (int c = 0; c < FIN; c += 256) {
        #pragma unroll
        for (int r = 0; r < 16; ++r) {
            xs[r][t] = xb[(size_t)r * FIN + c + t];
            gs[r][t] = wb[(size_t)r * FIN + c + t];
        }
        __syncthreads();
        #pragma unroll 8
        for (int kk = 0; kk < 256; ++kk)
            S += xs[i][kk] * gs[j][kk];
        #pragma unroll
        for (int u = 0; u < 16; ++u) {
            float v = xs[i][j * 16 + u]; nxp += v * v;
            float w = gs[j][i * 16 + u]; ngp += w * w;
        }
        __syncthreads();
    }

    // reduce norms
    red[i][j] = nxp;
    __syncthreads();
    if (t < 16) {
        float s = 0.f;
        for (int u = 0; u < 16; ++u) s += red[t][u];
        nX[t] = fmaxf(sqrtf(s), 1e-12f);
    }
    __syncthreads();
    red[j][i] = ngp;
    __syncthreads();
    if (t < 16) {
        float s = 0.f;
        for (int u = 0; u < 16; ++u) s += red[t][u];
        nG[t] = fmaxf(sqrtf(s), 1e-12f);
    }
    __syncthreads();

    // scores with diagonal boost (pre-softmax a_scores)
    float ev = (S / (nX[i] * nG[j])) * (1.f + (i == j ? aval : 0.f));
    EM[i][j] = ev;
    __syncthreads();

    // softmax over i (dim=1) for each column j
    if (t < 16) {
        float mx = -1e30f;
        for (int u = 0; u < 16; ++u) mx = fmaxf(mx, EM[u][t]);
        cmax[t] = mx;
    }
    __syncthreads();
    float p = expf(ev - cmax[j]);
    ESM[i][j] = p;
    __syncthreads();
    if (t < 16) {
        float s = 0.f;
        for (int u = 0; u < 16; ++u) s += ESM[u][t];
        csum[t] = s;
    }
    __syncthreads();
    float esm = p / csum[j];
    __syncthreads();
    ESM[i][j] = esm;
    __syncthreads();

    // from_i[i][k] = sum_t esm[i][t] * m[t][k]     (thread: row i, col k=j)
    float fi = 0.f;
    #pragma unroll
    for (int u = 0; u < 16; ++u) fi += ESM[i][u] * mb[u * 16 + j];
    FI[i][j] = fi;
    // from_j[r][k] = esm[r][k] * diag[k]           (row r=i, col k=j)
    FJ[i][j] = esm * mb[j * 17];
    __syncthreads();

    // e2[i][j] = sum_k from_i[i][k]*from_j[j][k]; then symmetrize
    float e2v = 0.f;
    #pragma unroll
    for (int k = 0; k < 16; ++k) e2v += FI[i][k] * FJ[j][k];
    red[i][j] = e2v;
    __syncthreads();
    float e2s = 0.5f * (red[i][j] + red[j][i]);
    e2_out[(size_t)b * 256 + i * 16 + j] = e2s;
    m_out [(size_t)b * 256 + i * 16 + j] = mb[i * 16 + j];

    // attn[i][f] = tanh(sum_j a_scores[j][i] * x[j][f]), overwrite wg region
    for (int c = 0; c < FIN; c += 256) {
        __syncthreads();
        #pragma unroll
        for (int r = 0; r < 16; ++r)
            xs[r][t] = xb[(size_t)r * FIN + c + t];
        __syncthreads();
        int i2 = t >> 4;
        int f0 = (t & 15) * 16;
        #pragma unroll
        for (int w = 0; w < 16; ++w) {
            int f = f0 + w;
            float s = 0.f;
            #pragma unroll
            for (int u = 0; u < 16; ++u) s += EM[u][i2] * xs[u][f];
            wb[(size_t)i2 * FIN + c + f] = tanhf(s);
        }
    }
}

// ---------------------------------------------------------------------------
// Kernel 3: out0[b][o] = sum_k attn[b][k] * W_mlp[o][k] + b_mlp[o],
//           out[b][t][o] = out0[b][o] for t = 0..15 (broadcast store).
// A: [1024,16384], W_mlp: [4096,16384]. Block tile 256(M) x 64(N), K-panel 16,
// double-buffered async LDS staging. 8 waves as 4(M)x2(N); 4x2 tiles per wave.
// ---------------------------------------------------------------------------
__global__ __launch_bounds__(256)
void k3_gemm_out(const float* __restrict__ A, const float* __restrict__ Wm,
                 const float* __restrict__ bm, float* __restrict__ Out)
{
    __shared__ float As[2][256][20];   // [buf][m][k]
    __shared__ float Bs[2][64][20];    // [buf][n][k]

    const int t    = threadIdx.x;
    const int lane = t & 31;
    const int wid  = t >> 5;
    const int wm   = wid >> 1;     // 0..3 : 64 rows each
    const int wn   = wid & 1;      // 0..1 : 32 cols each
    const int lm   = lane & 15;
    const int lh   = lane >> 4;
    const int iBase = blockIdx.y * 256;
    const int jBase = blockIdx.x * 64;

    v8f acc[4][2] = {};

    const int am  = (t >> 2);      // A: 4 chunks/thread (m, m+64, m+128, m+192)
    const int akq = (t & 3);

    // issue K-panel 0 into buffer 0 (5 async b128 per thread)
    #pragma unroll
    for (int rep = 0; rep < 4; ++rep) {
        int m = am + rep * 64;
        cp16(&A[(size_t)(iBase + m) * KBIG + akq * 4], &As[0][m][akq * 4]);
    }
    cp16(&Wm[(size_t)(jBase + (t >> 2)) * KBIG + (t & 3) * 4], &Bs[0][t >> 2][(t & 3) * 4]);

    int ib = 0;
    for (int kt = 0; kt < KBIG; kt += 16, ib ^= 1) {
        if (kt + 16 < KBIG) {
            #pragma unroll
            for (int rep = 0; rep < 4; ++rep) {
                int m = am + rep * 64;
                cp16(&A[(size_t)(iBase + m) * KBIG + kt + 16 + akq * 4],
                     &As[ib ^ 1][m][akq * 4]);
            }
            cp16(&Wm[(size_t)(jBase + (t >> 2)) * KBIG + kt + 16 + (t & 3) * 4],
                 &Bs[ib ^ 1][t >> 2][(t & 3) * 4]);
            // prefetch W_mlp two panels ahead into L2 (global_prefetch_b8)
            if (kt + 32 < KBIG)
                __builtin_prefetch(&Wm[(size_t)(jBase + (t >> 2)) * KBIG + kt + 32], 0, 1);
            WAIT_ASYNC(5);
        } else {
            WAIT_ASYNC(0);
        }
        __syncthreads();

        #pragma unroll
        for (int kk = 0; kk < 16; kk += 4) {
            v2f af[4], bf[2];
            #pragma unroll
            for (int mt = 0; mt < 4; ++mt)
                af[mt] = *(const v2f*)&As[ib][wm * 64 + mt * 16 + lm][kk + 2 * lh];
            #pragma unroll
            for (int nt = 0; nt < 2; ++nt)
                bf[nt] = *(const v2f*)&Bs[ib][wn * 32 + nt * 16 + lm][kk + 2 * lh];
            #pragma unroll
            for (int mt = 0; mt < 4; ++mt)
                #pragma unroll
                for (int nt = 0; nt < 2; ++nt)
                    acc[mt][nt] = __builtin_amdgcn_wmma_f32_16x16x4_f32(
                        false, af[mt], false, bf[nt],
                        (short)0, acc[mt][nt], false, false);
        }
        __syncthreads();
    }

    // epilogue: + bias, broadcast-store 16x along node dim
    #pragma unroll
    for (int mt = 0; mt < 4; ++mt)
        #pragma unroll
        for (int nt = 0; nt < 2; ++nt) {
            int col = jBase + wn * 32 + nt * 16 + lm;
            float bias = bm[col];
            #pragma unroll
            for (int r = 0; r < 8; ++r) {
                int row = iBase + wm * 64 + mt * 16 + 8 * lh + r;   // batch index
                float v = acc[mt][nt][r] + bias;
                float* o = Out + (size_t)row * NNODE * FOUT + col;
                #pragma unroll
                for (int rep = 0; rep < NNODE; ++rep)
                    o[(size_t)rep * FOUT] = v;
            }
        }
}

// ---------------------------------------------------------------------------
extern "C" void kernel_launch(void* const* d_in, const int* in_sizes, int n_in,
                              void* d_out, int out_size, void* d_ws, size_t ws_size,
                              hipStream_t stream)
{
    (void)in_sizes; (void)n_in; (void)out_size; (void)ws_size;

    const float* x   = (const float*)d_in[0];  // [B,16,1024]
    const float* mtx = (const float*)d_in[1];  // [B,16,16]
    const float* W1  = (const float*)d_in[2];  // [1024,1024]
    const float* b1  = (const float*)d_in[3];  // [1024]
    const float* Wm  = (const float*)d_in[4];  // [4096,16384]
    const float* bm  = (const float*)d_in[5];  // [4096]
    const float* ap  = (const float*)d_in[6];  // scalar

    float* out = (float*)d_out;                                   // [B,16,4096]
    float* e2  = out + (size_t)B_ * NNODE * FOUT;                 // [B,16,16]
    float* mo  = e2 + (size_t)B_ * NNODE * NNODE;                 // [B,16,16]

    float* wg = (float*)d_ws;  // 64 MiB: wg, then overwritten in-place by attn

    k1_gemm_wg<<<dim3(FIN / 64, MROWS / 128), 256, 0, stream>>>(x, W1, b1, wg);
    k2_attention<<<dim3(B_), 256, 0, stream>>>(x, mtx, ap, wg, e2, mo);
    k3_gemm_out<<<dim3(FOUT / 64, B_ / 256), 256, 0, stream>>>(wg, Wm, bm, out);
}